// SAGE_23210003267951
// MI455X (gfx1250) — compile-verified
//
#include <hip/hip_runtime.h>

#define N_NODES 50000
#define N_EDGES 800000
#define D 128

typedef __attribute__((ext_vector_type(2))) float v2f;
typedef __attribute__((ext_vector_type(8))) float v8f;

// ---------------------------------------------------------------------------
// Zero a float buffer (agg accumulator / degree buffer).
// ---------------------------------------------------------------------------
__global__ void sage_zero_f32(float* __restrict__ p, int n) {
    int i = blockIdx.x * blockDim.x + threadIdx.x;
    if (i < n) p[i] = 0.0f;
}

// ---------------------------------------------------------------------------
// Per-destination degree via hardware float atomics (computed once, the graph
// is identical for both layers).
// ---------------------------------------------------------------------------
__global__ void sage_degree(const long long* __restrict__ eidx,
                            float* __restrict__ deg) {
    int e = blockIdx.x * blockDim.x + threadIdx.x;
    if (e < N_EDGES) {
        int dst = (int)eidx[N_EDGES + e];
        unsafeAtomicAdd(deg + dst, 1.0f);
    }
}

__global__ void sage_recip_clamp(float* __restrict__ deg) {
    int i = blockIdx.x * blockDim.x + threadIdx.x;
    if (i < N_NODES) deg[i] = 1.0f / fmaxf(deg[i], 1.0f);
}

// ---------------------------------------------------------------------------
// Scatter-add of source rows into agg[dst]. One wave (32 lanes) per edge,
// float4 gather per lane + 4 global_atomic_add_f32. Both the feature table
// (25.6 MB) and agg (25.6 MB) are L2-resident on MI455X (192 MB L2).
// 256 threads = 8 waves = 8 edges per block; 800000/8 = 100000 blocks exact.
// ---------------------------------------------------------------------------
__global__ void __launch_bounds__(256) sage_scatter(
    const float* __restrict__ x,
    const long long* __restrict__ eidx,
    float* __restrict__ agg) {
    int e    = blockIdx.x * 8 + (threadIdx.x >> 5);
    int lane = threadIdx.x & 31;
    long long src = eidx[e];
    long long dst = eidx[N_EDGES + e];
    const float4 v = ((const float4*)(x + src * D))[lane];
    float* ar = agg + dst * D + lane * 4;
    unsafeAtomicAdd(ar + 0, v.x);
    unsafeAtomicAdd(ar + 1, v.y);
    unsafeAtomicAdd(ar + 2, v.z);
    unsafeAtomicAdd(ar + 3, v.w);
}

// ---------------------------------------------------------------------------
// Fused SAGE layer GEMM:
//   out = relu( (agg * invdeg) @ Wn + xin @ Wr + bias )
// Block = 256 threads = 8 waves; one block owns a 16-row stripe (50000/16 =
// 3125 blocks exact, no divergence -> EXEC all-1s as WMMA requires).
// A-tiles (16x128, normalized agg and root features) are staged in LDS once
// per block; each wave computes a 16x16 output tile with fp32 WMMA
// (v_wmma_f32_16x16x4_f32), K swept in steps of 4, both matmuls accumulated
// into one v8f fragment.
//
// Fragment layouts (ISA 7.12.2):
//   A 16x4 f32 : lane L -> row L%16, VGPR0/1 = K = (L/16)*2, (L/16)*2+1
//   B 4x16 f32 : lane L -> col L%16, VGPR0/1 = same K pair
//   C/D 16x16  : VGPR v -> row (v + (L/16)*8), col L%16
// ---------------------------------------------------------------------------
__global__ void __launch_bounds__(256) sage_gemm(
    const float* __restrict__ agg,
    const float* __restrict__ xin,
    const float* __restrict__ invdeg,
    const float* __restrict__ Wn,
    const float* __restrict__ Wr,
    const float* __restrict__ bias,
    float* __restrict__ out) {
    __shared__ float sA[16 * D];   // normalized neighbor aggregate tile
    __shared__ float sX[16 * D];   // root feature tile

    const int t  = threadIdx.x;
    const int m0 = blockIdx.x * 16;

    // ---- stage A-tiles: 256 threads x 8 floats each per matrix -----------
    {
        const int r  = t >> 4;          // 0..15 tile row
        const int c0 = (t & 15) * 8;    // 0..120, 32B aligned
        const float inv = invdeg[m0 + r];
        const float4* ga = (const float4*)(agg + (size_t)(m0 + r) * D + c0);
        const float4* gx = (const float4*)(xin + (size_t)(m0 + r) * D + c0);
        float4 a0 = ga[0], a1 = ga[1];
        float4 x0 = gx[0], x1 = gx[1];
        float4* la = (float4*)(sA + r * D + c0);
        float4* lx = (float4*)(sX + r * D + c0);
        la[0] = make_float4(a0.x * inv, a0.y * inv, a0.z * inv, a0.w * inv);
        la[1] = make_float4(a1.x * inv, a1.y * inv, a1.z * inv, a1.w * inv);
        lx[0] = x0;
        lx[1] = x1;
    }
    __syncthreads();

    const int lane = t & 31;
    const int wv   = t >> 5;
    const int n0   = wv * 16;        // this wave's 16-column tile
    const int col  = lane & 15;      // A row / B col / D col for this lane
    const int half = lane >> 4;      // K sub-pair selector

    v8f acc = {};
#pragma unroll
    for (int k = 0; k < D; k += 4) {
        const int ka = k + half * 2;
        v2f aN = *(const v2f*)(sA + col * D + ka);            // ds_load_b64
        v2f aR = *(const v2f*)(sX + col * D + ka);
        v2f bN = { Wn[ka * D + n0 + col], Wn[(ka + 1) * D + n0 + col] };
        v2f bR = { Wr[ka * D + n0 + col], Wr[(ka + 1) * D + n0 + col] };
        acc = __builtin_amdgcn_wmma_f32_16x16x4_f32(
            false, aN, false, bN, (short)0, acc, false, false);
        acc = __builtin_amdgcn_wmma_f32_16x16x4_f32(
            false, aR, false, bR, (short)0, acc, false, false);
    }

    // ---- epilogue: bias + ReLU, coalesced 16-lane stores -----------------
    const float bv = bias[n0 + col];
#pragma unroll
    for (int v = 0; v < 8; v++) {
        const int orow = m0 + v + half * 8;
        out[(size_t)orow * D + n0 + col] = fmaxf(acc[v] + bv, 0.0f);
    }
}

// ---------------------------------------------------------------------------
// Host launcher. d_in order: x, edge_index(int64), Wn1, Wr1, b1, Wn2, Wr2, b2.
// Workspace layout: agg[N*D] | hid[N*D] | invdeg[N]  (~51.5 MB).
// ---------------------------------------------------------------------------
extern "C" void kernel_launch(void* const* d_in, const int* in_sizes, int n_in,
                              void* d_out, int out_size, void* d_ws,
                              size_t ws_size, hipStream_t stream) {
    const float*     x    = (const float*)d_in[0];
    const long long* eidx = (const long long*)d_in[1];
    const float*     Wn1  = (const float*)d_in[2];
    const float*     Wr1  = (const float*)d_in[3];
    const float*     b1   = (const float*)d_in[4];
    const float*     Wn2  = (const float*)d_in[5];
    const float*     Wr2  = (const float*)d_in[6];
    const float*     b2   = (const float*)d_in[7];
    float*           out  = (float*)d_out;

    float* agg    = (float*)d_ws;
    float* hid    = agg + (size_t)N_NODES * D;
    float* invdeg = hid + (size_t)N_NODES * D;

    const int ND = N_NODES * D;

    // degrees -> 1/max(deg,1), computed once (graph shared by both layers)
    sage_zero_f32<<<(N_NODES + 255) / 256, 256, 0, stream>>>(invdeg, N_NODES);
    sage_degree<<<(N_EDGES + 255) / 256, 256, 0, stream>>>(eidx, invdeg);
    sage_recip_clamp<<<(N_NODES + 255) / 256, 256, 0, stream>>>(invdeg);

    // ---- layer 1 ----
    sage_zero_f32<<<(ND + 255) / 256, 256, 0, stream>>>(agg, ND);
    sage_scatter<<<N_EDGES / 8, 256, 0, stream>>>(x, eidx, agg);
    sage_gemm<<<N_NODES / 16, 256, 0, stream>>>(agg, x, invdeg, Wn1, Wr1, b1,
                                                hid);

    // ---- layer 2 ----
    sage_zero_f32<<<(ND + 255) / 256, 256, 0, stream>>>(agg, ND);
    sage_scatter<<<N_EDGES / 8, 256, 0, stream>>>(hid, eidx, agg);
    sage_gemm<<<N_NODES / 16, 256, 0, stream>>>(agg, hid, invdeg, Wn2, Wr2, b2,
                                                out);
}